// GraphSAGEEncoder_66855460929638
// MI455X (gfx1250) — compile-verified
//
#include <hip/hip_runtime.h>
#include <hip/hip_bf16.h>

#define NN 100000
#define NE 1200000

typedef __attribute__((ext_vector_type(2))) float v2f;
typedef __attribute__((ext_vector_type(8))) float v8f;

// ---------------------------------------------------------------- utilities
// n4 = count in float4 units
__global__ void zero_kernel(float4* __restrict__ p, int n4) {
    int i = blockIdx.x * blockDim.x + threadIdx.x;
    int stride = gridDim.x * blockDim.x;
    const float4 z = {0.f, 0.f, 0.f, 0.f};
    for (; i < n4; i += stride) p[i] = z;
}

// 16 threads per edge; each thread gathers a float4 (b128) and does 4 atomics
__global__ void scatter_mean_accum(const float* __restrict__ X,
                                   const int* __restrict__ src,
                                   const int* __restrict__ dst,
                                   float* __restrict__ agg,
                                   float* __restrict__ cnt) {
    unsigned t = blockIdx.x * blockDim.x + threadIdx.x;
    const unsigned total = (unsigned)NE * 16u;
    if (t >= total) return;
    unsigned e = t >> 4;
    unsigned q = (t & 15u) << 2;        // dim offset: 0,4,...,60
    int s  = src[e];
    int dd = dst[e];
    const float4 v = *(const float4*)(X + (size_t)s * 64 + q);
    float* a = agg + (size_t)dd * 64 + q;
    atomicAdd(a + 0, v.x);
    atomicAdd(a + 1, v.y);
    atomicAdd(a + 2, v.z);
    atomicAdd(a + 3, v.w);
    if (cnt != nullptr && (t & 15u) == 0) atomicAdd(&cnt[dd], 1.0f);
}

// float4-vectorized mean divide: i ranges over nnodes*16 float4's
__global__ void mean_div(float4* __restrict__ agg, const float* __restrict__ cnt,
                         int nnodes) {
    int i = blockIdx.x * blockDim.x + threadIdx.x;
    if (i >= nnodes * 16) return;
    const float inv = 1.0f / fmaxf(cnt[i >> 4], 1.0f);
    float4 v = agg[i];
    v.x *= inv; v.y *= inv; v.z *= inv; v.w *= inv;
    agg[i] = v;
}

// ------------------------------------------------- WMMA dense SAGE layer
// Out[N, OUT] = act( Xa @ Wl + bias + Xr @ Wr ),  K fixed at 64.
// V_WMMA_F32_16X16X4_F32; each wave32 owns TILES consecutive 16-row tiles and
// caches the B (weight) fragments in registers across them.
//
// ISA VGPR layouts (CDNA5 ISA):
//   A 16x4 f32 : lane&15 = M row, (lane>>4)*2 = K base; {v0,v1} = K, K+1
//   B  4x16 f32: lane&15 = N col, (lane>>4)*2 = K base; {v0,v1} = K, K+1
//   C/D 16x16  : lane&15 = N col, VGPR v -> M = v + 8*(lane>>4)
template <int OUT, bool RELU, int TILES>
__global__ __launch_bounds__(128) void sage_dense_wmma(
    const float* __restrict__ Xa,   // [N, 64] mean-aggregated neighbors
    const float* __restrict__ Xr,   // [N, 64] root features
    const float* __restrict__ Wl,   // [64, OUT]
    const float* __restrict__ bias, // [OUT]
    const float* __restrict__ Wr,   // [64, OUT]
    float* __restrict__ Out,        // [N, OUT]
    int nrows) {
    const int lane = threadIdx.x & 31;
    const int wave = blockIdx.x * (blockDim.x >> 5) + (threadIdx.x >> 5);
    const int base = wave * (16 * TILES);
    if (base >= nrows) return;          // wave-uniform: EXEC all-1s for WMMA

    const int mrow  = lane & 15;
    const int khalf = (lane >> 4) << 1;   // 0 or 2
    const int mbase = (lane >> 4) << 3;   // C/D: M = v + mbase

#pragma unroll
    for (int nt = 0; nt < OUT / 16; ++nt) {
        const int col = nt * 16 + mrow;

        // cache B fragments for this n-tile across all row tiles
        v2f bl[16], br[16];
#pragma unroll
        for (int i = 0; i < 16; ++i) {
            const int k = i * 4 + khalf;
            bl[i].x = Wl[k * OUT + col];
            bl[i].y = Wl[(k + 1) * OUT + col];
            br[i].x = Wr[k * OUT + col];
            br[i].y = Wr[(k + 1) * OUT + col];
        }
        const float bv = bias[col];

#pragma unroll
        for (int t = 0; t < TILES; ++t) {
            const int r0 = base + t * 16;
            if (r0 >= nrows) break;     // uniform across wave
            int row = r0 + mrow;
            if (row >= nrows) row = nrows - 1;  // clamped load; store predicated
            const float* xa = Xa + (size_t)row * 64 + khalf;
            const float* xr = Xr + (size_t)row * 64 + khalf;

            v8f accL;                   // lin_l chain, seeded with bias
            v8f accR = {};              // lin_r chain
#pragma unroll
            for (int i = 0; i < 8; ++i) accL[i] = bv;

#pragma unroll
            for (int i = 0; i < 16; ++i) {
                const v2f a  = *(const v2f*)(xa + i * 4);   // b64, 8B-aligned
                accL = __builtin_amdgcn_wmma_f32_16x16x4_f32(
                    false, a, false, bl[i], (short)0, accL, false, false);
                const v2f a2 = *(const v2f*)(xr + i * 4);
                accR = __builtin_amdgcn_wmma_f32_16x16x4_f32(
                    false, a2, false, br[i], (short)0, accR, false, false);
            }

#pragma unroll
            for (int v = 0; v < 8; ++v) {
                float val = accL[v] + accR[v];
                if (RELU) val = fmaxf(val, 0.0f);
                const int orow = r0 + mbase + v;
                if (orow < nrows) Out[(size_t)orow * OUT + col] = val;
            }
        }
    }
}

// ------------------------------------------------------------------ launch
extern "C" void kernel_launch(void* const* d_in, const int* in_sizes, int n_in,
                              void* d_out, int out_size, void* d_ws, size_t ws_size,
                              hipStream_t stream) {
    const float* x   = (const float*)d_in[0];
    const int*   ei  = (const int*)d_in[1];   // [2, NE] int32
    const float* W1l = (const float*)d_in[2];
    const float* b1  = (const float*)d_in[3];
    const float* W1r = (const float*)d_in[4];
    const float* W2l = (const float*)d_in[5];
    const float* b2  = (const float*)d_in[6];
    const float* W2r = (const float*)d_in[7];
    float* out = (float*)d_out;

    const int* src = ei;
    const int* dst = ei + NE;

    // workspace layout: agg[N*64] | cnt[N] | h[N*64]   (~52 MB)
    float* agg = (float*)d_ws;
    float* cnt = agg + (size_t)NN * 64;
    float* h   = cnt + NN;

    constexpr int TILES = 4;
    const unsigned sc_total = (unsigned)NE * 16u;
    const int sc_blocks     = (int)((sc_total + 255u) / 256u);
    const int md_blocks     = (NN * 16 + 255) / 256;
    const int tiles         = (NN + 15) / 16;
    const int waves         = (tiles + TILES - 1) / TILES;
    const int dense_blocks  = (waves + 3) / 4;   // 4 waves / block

    // -------- layer 1: agg = mean_{dst}(x[src]); h = relu(agg@W1l + b1 + x@W1r)
    zero_kernel<<<1024, 256, 0, stream>>>((float4*)agg, NN * 65 / 4); // agg+cnt
    scatter_mean_accum<<<sc_blocks, 256, 0, stream>>>(x, src, dst, agg, cnt);
    mean_div<<<md_blocks, 256, 0, stream>>>((float4*)agg, cnt, NN);
    sage_dense_wmma<64, true, TILES><<<dense_blocks, 128, 0, stream>>>(
        agg, x, W1l, b1, W1r, h, NN);

    // -------- layer 2: agg = mean_{dst}(h[src]); out = agg@W2l + b2 + h@W2r
    zero_kernel<<<1024, 256, 0, stream>>>((float4*)agg, NN * 16);
    scatter_mean_accum<<<sc_blocks, 256, 0, stream>>>(h, src, dst, agg, nullptr);
    mean_div<<<md_blocks, 256, 0, stream>>>((float4*)agg, cnt, NN);
    sage_dense_wmma<32, false, TILES><<<dense_blocks, 128, 0, stream>>>(
        agg, h, W2l, b2, W2r, out, NN);
}